// OLNRPN_47047071760588
// MI455X (gfx1250) — compile-verified
//
#include <hip/hip_runtime.h>

// ---------------------------------------------------------------------------
// OLN-RPN proposal generation for MI455X (gfx1250, wave32).
//   Stage 1: per-(image, FPN level) exact top-2000 via 4-pass radix threshold
//            (LDS histograms) + deterministic prefix-scan compaction, fused
//            with linear box decode + clip + empty-box filter.
//   Stage 2: per-image greedy NMS, 1000 sequential steps; candidates live in
//            registers, argmax via wave32 shfl_xor + cross-wave LDS reduce.
//            Candidate scores are DMA'd into LDS by the Tensor Data Mover
//            (tensor_load_to_lds + s_wait_tensorcnt), overlapping with the
//            per-thread global b128 box loads.
// No GEMM content -> no WMMA (top-k/NMS is min/max/select arithmetic).
// ---------------------------------------------------------------------------

#define NEGV (-1e9f)

constexpr int   Bn      = 16;
constexpr int   NLVL    = 4;
constexpr int   K       = 2000;          // pre-NMS per level
constexpr int   LSTRIDE = 2048;          // per-level slot stride (48 pad slots)
constexpr int   NSLOT   = NLVL * LSTRIDE;// 8192 candidate slots per image
constexpr int   POST    = 1000;
constexpr float IMGMAX  = 1024.0f;
constexpr float NMS_T   = 0.7f;

typedef unsigned int u32x4 __attribute__((ext_vector_type(4)));
typedef int          i32x4 __attribute__((ext_vector_type(4)));
typedef int          i32x8 __attribute__((ext_vector_type(8)));

#if __has_builtin(__builtin_amdgcn_tensor_load_to_lds) && \
    __has_builtin(__builtin_amdgcn_s_wait_tensorcnt)
#define OLN_USE_TDM 1
#else
#define OLN_USE_TDM 0
#endif

__device__ __forceinline__ unsigned fkey(float f) {
  // monotonic float -> uint mapping (ascending)
  unsigned u = __float_as_uint(f);
  return u ^ ((u >> 31) ? 0xFFFFFFFFu : 0x80000000u);
}

// ---------------------------------------------------------------------------
// Stage 1: top-K selection + box decode.  grid = Bn blocks, 1024 threads.
// ---------------------------------------------------------------------------
__global__ __launch_bounds__(1024)
void olnrpn_topk_kernel(const float* __restrict__ anc,    // [A,4]
                        const float* __restrict__ del,    // [B,A,4]
                        const float* __restrict__ lg,     // [B,A]
                        float4* __restrict__ cand_boxes,  // [B,NSLOT]
                        float*  __restrict__ cand_scores, // [B,NSLOT]
                        int A, int lv) {
  const int b   = blockIdx.x;
  const int tid = threadIdx.x;
  const int lane = tid & 31, wave = tid >> 5;

  const float*  logits = lg + (size_t)b * A;
  const float4* deltas = (const float4*)del + (size_t)b * A;
  const float4* anchors = (const float4*)anc;
  float4* obox = cand_boxes  + (size_t)b * NSLOT + lv * LSTRIDE;
  float*  osc  = cand_scores + (size_t)b * NSLOT + lv * LSTRIDE;

  __shared__ unsigned hist[256];
  __shared__ unsigned shWG[32], shWE[32];
  __shared__ unsigned sh_prefix, sh_R;

  // init pad slots (deterministic every call)
  for (int i = K + tid; i < LSTRIDE; i += 1024) {
    osc[i]  = NEGV;
    obox[i] = make_float4(0.f, 0.f, 0.f, 0.f);
  }
  if (tid == 0) { sh_prefix = 0u; sh_R = (unsigned)K; }
  __syncthreads();

  // ---- 4-pass MSB-first radix threshold search -----------------------------
  for (int p = 3; p >= 0; --p) {
    for (int i = tid; i < 256; i += 1024) hist[i] = 0u;
    __syncthreads();
    const unsigned prefix = sh_prefix;
    const int shift = p * 8;
    const int hiShift = shift + 8;
    for (int i = tid; i < A; i += 1024) {
#if defined(__gfx1250__)
      if (i + 8192 < A) __builtin_prefetch(&logits[i + 8192], 0, 1);
#endif
      unsigned k = fkey(logits[i]);
      bool match = (p == 3) || ((k >> hiShift) == (prefix >> hiShift));
      if (match) atomicAdd(&hist[(k >> shift) & 255u], 1u);
    }
    __syncthreads();
    if (tid == 0) {
      unsigned R = sh_R, acc = 0u;
      int chosen = 0;
      for (int bin = 255; bin >= 0; --bin) {
        unsigned h = hist[bin];
        if (acc + h >= R) { chosen = bin; R -= acc; break; }
        acc += h;
      }
      sh_prefix = prefix | ((unsigned)chosen << shift);
      sh_R = R;
    }
    __syncthreads();
  }
  const unsigned T     = sh_prefix;     // exact K-th largest key
  const unsigned quota = sh_R;          // #(== T) elements to admit
  // #(> T) == K - quota, guaranteed.

  // ---- deterministic compaction: chunked counts + block prefix scan --------
  const int chunk = (A + 1023) >> 10;
  const int begin = tid * chunk;
  const int end   = min(begin + chunk, A);

  unsigned g = 0u, e = 0u;
  for (int i = begin; i < end; ++i) {
    unsigned k = fkey(logits[i]);
    g += (k > T);
    e += (k == T);
  }
  // wave32 inclusive scan
  unsigned gi = g, ei = e;
  #pragma unroll
  for (int off = 1; off < 32; off <<= 1) {
    unsigned tg = __shfl_up(gi, off, 32);
    unsigned te = __shfl_up(ei, off, 32);
    if (lane >= off) { gi += tg; ei += te; }
  }
  if (lane == 31) { shWG[wave] = gi; shWE[wave] = ei; }
  __syncthreads();
  if (wave == 0) {
    unsigned a = shWG[lane], c = shWE[lane];
    unsigned ai = a, ci = c;
    #pragma unroll
    for (int off = 1; off < 32; off <<= 1) {
      unsigned ta = __shfl_up(ai, off, 32);
      unsigned tc = __shfl_up(ci, off, 32);
      if (lane >= off) { ai += ta; ci += tc; }
    }
    shWG[lane] = ai - a;   // exclusive offsets across waves
    shWE[lane] = ci - c;
  }
  __syncthreads();
  unsigned gpos = shWG[wave] + (gi - g);  // global exclusive rank among (>T)
  unsigned epos = shWE[wave] + (ei - e);  // global exclusive rank among (==T)

  // ---- write pass: decode boxes for admitted elements ----------------------
  for (int i = begin; i < end; ++i) {
    float s = logits[i];
    unsigned k = fkey(s);
    int pos = -1;
    if (k > T) {
      pos = (int)gpos++;
    } else if (k == T) {
      if (epos < quota) pos = (int)(K - quota + epos);
      epos++;
    }
    if (pos >= 0) {
      float4 a4 = anchors[i];
      float4 d4 = deltas[i];
      float w  = a4.z - a4.x, h = a4.w - a4.y;
      float cx = 0.5f * (a4.x + a4.z), cy = 0.5f * (a4.y + a4.w);
      float l  = fmaxf(d4.x, 0.f) * w;
      float t  = fmaxf(d4.y, 0.f) * h;
      float r  = fmaxf(d4.z, 0.f) * w;
      float bo = fmaxf(d4.w, 0.f) * h;
      float x1 = fminf(fmaxf(cx - l,  0.f), IMGMAX);
      float y1 = fminf(fmaxf(cy - t,  0.f), IMGMAX);
      float x2 = fminf(fmaxf(cx + r,  0.f), IMGMAX);
      float y2 = fminf(fmaxf(cy + bo, 0.f), IMGMAX);
      obox[pos] = make_float4(x1, y1, x2, y2);
      osc[pos]  = ((x2 > x1) && (y2 > y1)) ? s : NEGV;
    }
  }
}

// ---------------------------------------------------------------------------
// Stage 2: greedy NMS.  grid = Bn blocks, 1024 threads (8 slots per thread,
// all candidate state in registers; compile-time indices only).
// Scores are staged LDS via the Tensor Data Mover, overlapping the box loads.
// ---------------------------------------------------------------------------
__global__ __launch_bounds__(1024)
void olnrpn_nms_kernel(const float4* __restrict__ cand_boxes,
                       const float*  __restrict__ cand_scores,
                       float* __restrict__ out) {          // [B,POST,5]
  const int b   = blockIdx.x;
  const int tid = threadIdx.x;
  const int lane = tid & 31, wave = tid >> 5;

  const float4* boxes  = cand_boxes  + (size_t)b * NSLOT;
  const float*  scores = cand_scores + (size_t)b * NSLOT;
  float* o = out + (size_t)b * POST * 5;

#if OLN_USE_TDM
  __shared__ float lds_scores[NSLOT];                      // 32 KB
  if (wave == 0) {
    // ---- TDM descriptor: 1 row of NSLOT f32 elements, global -> LDS ----
    const unsigned lds_base =
        (unsigned)(uintptr_t)(&lds_scores[0]);             // low 32b = LDS offset
    const unsigned long long ga = (unsigned long long)(uintptr_t)scores;
    u32x4 g0;
    g0[0] = 0x1u;                                  // count=1, user descriptor
    g0[1] = lds_base;                              // lds_addr (bytes)
    g0[2] = (unsigned)(ga & 0xFFFFFFFFu);          // global_addr[31:0]
    g0[3] = (unsigned)((ga >> 32) & 0x01FFFFFFu)   // global_addr[56:32]
            | 0x80000000u;                         // type=2 ("image")
    i32x8 g1;
    g1[0] = 0x20000;                               // data_size=2 (4B), mask=0
    g1[1] = (int)((NSLOT & 0xFFFF) << 16);         // tensor_dim0[15:0]
    g1[2] = (int)((NSLOT >> 16) & 0xFFFF);         // tensor_dim0[31:16]
    g1[3] = (int)((NSLOT & 0xFFFF) << 16);         // tile_dim0 = NSLOT
    g1[4] = 0;                                     // tile_dim1/2 = 0 (1D)
    g1[5] = (int)NSLOT;                            // tensor_dim0_stride
    g1[6] = 0;
    g1[7] = 0;
    i32x4 z4 = {0, 0, 0, 0};
#if __has_include(<hip/amd_detail/amd_gfx1250_TDM.h>)
    i32x8 z8 = {0, 0, 0, 0, 0, 0, 0, 0};
    __builtin_amdgcn_tensor_load_to_lds(g0, g1, z4, z4, z8, 0);  // 6-arg lane
#else
    __builtin_amdgcn_tensor_load_to_lds(g0, g1, z4, z4, 0);      // 5-arg lane
#endif
  }
#endif

  // Per-thread box loads (global b128) proceed concurrently with the TDM DMA.
  float sx1[8], sy1[8], sx2[8], sy2[8], sar[8], ssc[8];
  int   slvl[8];
  #pragma unroll
  for (int j = 0; j < 8; ++j) {
    int slot = tid + j * 1024;
    float4 bx = boxes[slot];
    sx1[j] = bx.x; sy1[j] = bx.y; sx2[j] = bx.z; sy2[j] = bx.w;
    sar[j] = fmaxf(bx.z - bx.x, 0.f) * fmaxf(bx.w - bx.y, 0.f);
    slvl[j] = slot >> 11;                 // LSTRIDE == 2048
  }

#if OLN_USE_TDM
  if (wave == 0) __builtin_amdgcn_s_wait_tensorcnt(0);
  __syncthreads();                        // scores now visible in LDS
  #pragma unroll
  for (int j = 0; j < 8; ++j) ssc[j] = lds_scores[tid + j * 1024];
#else
  #pragma unroll
  for (int j = 0; j < 8; ++j) ssc[j] = scores[tid + j * 1024];
#endif

  __shared__ float wsc[32];
  __shared__ int   wid[32];
  __shared__ float bb_x1, bb_y1, bb_x2, bb_y2, bb_ar, bb_sc;
  __shared__ int   bb_lvl, bb_slot;

  for (int it = 0; it < POST; ++it) {
    // ---- local argmax (lowest-index tie-break, matches jnp.argmax) ----
    float bs = -INFINITY; int bi = NSLOT;
    #pragma unroll
    for (int j = 0; j < 8; ++j) {
      int slot = tid + j * 1024;
      float s = ssc[j];
      if (s > bs || (s == bs && slot < bi)) { bs = s; bi = slot; }
    }
    // ---- wave32 reduce ----
    #pragma unroll
    for (int off = 16; off > 0; off >>= 1) {
      float os = __shfl_xor(bs, off, 32);
      int   oi = __shfl_xor(bi, off, 32);
      if (os > bs || (os == bs && oi < bi)) { bs = os; bi = oi; }
    }
    if (lane == 0) { wsc[wave] = bs; wid[wave] = bi; }
    __syncthreads();                                   // (1)
    if (wave == 0) {
      bs = wsc[lane]; bi = wid[lane];
      #pragma unroll
      for (int off = 16; off > 0; off >>= 1) {
        float os = __shfl_xor(bs, off, 32);
        int   oi = __shfl_xor(bi, off, 32);
        if (os > bs || (os == bs && oi < bi)) { bs = os; bi = oi; }
      }
      if (lane == 0) { bb_slot = bi; bb_sc = bs; }
    }
    __syncthreads();                                   // (2)
    const int bslot = bb_slot;

    // ---- owner publishes winner, writes output row, self-suppresses ----
    #pragma unroll
    for (int j = 0; j < 8; ++j) {
      if (bslot == tid + j * 1024) {
        bb_x1 = sx1[j]; bb_y1 = sy1[j]; bb_x2 = sx2[j]; bb_y2 = sy2[j];
        bb_ar = sar[j]; bb_lvl = slvl[j];
        float* row = o + (size_t)it * 5;
        row[0] = sx1[j]; row[1] = sy1[j]; row[2] = sx2[j]; row[3] = sy2[j];
        row[4] = bb_sc;
        ssc[j] = NEGV;                     // idx == j suppression
      }
    }
    __syncthreads();                                   // (3)

    // ---- IoU suppression within same FPN level ----
    const float jx1 = bb_x1, jy1 = bb_y1, jx2 = bb_x2, jy2 = bb_y2;
    const float jar = bb_ar;
    const int   jl  = bb_lvl;
    #pragma unroll
    for (int j = 0; j < 8; ++j) {
      float iw = fminf(jx2, sx2[j]) - fmaxf(jx1, sx1[j]);
      float ih = fminf(jy2, sy2[j]) - fmaxf(jy1, sy1[j]);
      iw = fmaxf(iw, 0.f); ih = fmaxf(ih, 0.f);
      float inter = iw * ih;
      float iou = inter / fmaxf(sar[j] + jar - inter, 1e-6f);
      if (iou > NMS_T && slvl[j] == jl) ssc[j] = NEGV;
    }
    // next iteration's barrier (1) orders suppression vs. shared rewrites
  }
}

// ---------------------------------------------------------------------------
extern "C" void kernel_launch(void* const* d_in, const int* in_sizes, int n_in,
                              void* d_out, int out_size, void* d_ws, size_t ws_size,
                              hipStream_t stream) {
  (void)n_in; (void)out_size; (void)ws_size;

  // d_in order (setup_inputs dict order):
  //   [anchors_p2, deltas_p2, logits_p2, anchors_p3, ..., logits_p5]
  float4* cand_boxes  = (float4*)d_ws;                                   // 2 MB
  float*  cand_scores = (float*)((char*)d_ws +
                                 (size_t)Bn * NSLOT * sizeof(float4));   // 512 KB

  for (int lv = 0; lv < NLVL; ++lv) {
    const float* anc = (const float*)d_in[3 * lv + 0];
    const float* del = (const float*)d_in[3 * lv + 1];
    const float* lg  = (const float*)d_in[3 * lv + 2];
    const int A = in_sizes[3 * lv] / 4;   // anchors are [A,4] flat
    olnrpn_topk_kernel<<<dim3(Bn), dim3(1024), 0, stream>>>(
        anc, del, lg, cand_boxes, cand_scores, A, lv);
  }
  olnrpn_nms_kernel<<<dim3(Bn), dim3(1024), 0, stream>>>(
      cand_boxes, cand_scores, (float*)d_out);
}